// TaskAlignedAssigner_26542897889477
// MI455X (gfx1250) — compile-verified
//
#include <hip/hip_runtime.h>
#include <cstdint>

// ---------------------------------------------------------------------------
// TaskAlignedAssigner for gfx1250 (MI455X).
//   B=32, N=64, HW=8400, C=80, TOPK=13, ALPHA=1, BETA=6
// Dense intermediates OV/AM stored (B,HW,N) (N contiguous) in d_ws.
// bbox_scores gather implemented as one-hot GEMM on V_WMMA_F32_16X16X4_F32.
// Required ws: 2*B*HW*N*4 + ~6.5MB  ~= 145 MB.
// ---------------------------------------------------------------------------

#define B_   32
#define N_   64
#define HW_  8400
#define C_   80
#define TK_  13
#define EPSF 1e-9f

typedef float v2f __attribute__((ext_vector_type(2)));
typedef float v8f __attribute__((ext_vector_type(8)));

// ---------------------------------------------------------------------------
// Zero the scatter state (must run every call; harness does not re-poison).
// ---------------------------------------------------------------------------
__global__ void k_init(unsigned* __restrict__ cnt, int* __restrict__ seln,
                       unsigned* __restrict__ pal, unsigned* __restrict__ pov) {
    int t = blockIdx.x * blockDim.x + threadIdx.x;
    if (t < B_ * HW_) { cnt[t] = 0u; seln[t] = 0; }
    if (t < B_ * N_)  { pal[t] = 0u; pov[t]  = 0u; }
}

// ---------------------------------------------------------------------------
// Pairwise kernel: one wave per (b, 16-anchor tile).
//  - scores for all 64 gts via one-hot WMMA (f32, exact)
//  - fused in-gts mask, IoU, align metric; masked stores to OV/AM (B,HW,N)
// ---------------------------------------------------------------------------
__global__ __launch_bounds__(32)
void k_metric(const float* __restrict__ pd_scores,  // (B,HW,C)
              const float* __restrict__ pd_bboxes,  // (B,HW,4)
              const float* __restrict__ anc,        // (HW,2)
              const int*   __restrict__ gt_labels,  // (B,N)
              const float* __restrict__ gt_bboxes,  // (B,N,4)
              const float* __restrict__ mask_gt,    // (B,N)
              float* __restrict__ OV,               // (B,HW,N)
              float* __restrict__ AM) {             // (B,HW,N)
    const int lane = threadIdx.x;
    const int b    = blockIdx.y;
    const int hw0  = blockIdx.x * 16;

    // A-matrix (16 anchors x 4 classes per WMMA) lane layout:
    // lanes 0-15: M=lane, K={0,1}; lanes 16-31: M=lane-16, K={2,3}
    const int mrow = lane & 15;
    const int kb   = (lane >> 4) * 2;
    const float* srow = pd_scores + ((size_t)b * HW_ + (hw0 + mrow)) * C_;
    v2f a[20];
#pragma unroll
    for (int k = 0; k < 20; ++k)
        a[k] = *(const v2f*)(srow + k * 4 + kb);

    // D-matrix (16x16 f32) lane layout: this lane owns column N = lane&15,
    // rows M = r + 8*(lane>=16) for VGPR r in 0..7.
    const int hwl = hw0 + ((lane >> 4) << 3);

    for (int nt = 0; nt < 4; ++nt) {
        const int n   = nt * 16 + (lane & 15);
        const int lbl = gt_labels[b * N_ + n];

        v8f acc = {};
#pragma unroll
        for (int k = 0; k < 20; ++k) {
            // B-matrix (4 classes x 16 gts): one-hot of this gt's label.
            const int c0 = k * 4 + kb;
            v2f bb;
            bb.x = (lbl == c0)     ? 1.0f : 0.0f;
            bb.y = (lbl == c0 + 1) ? 1.0f : 0.0f;
            acc = __builtin_amdgcn_wmma_f32_16x16x4_f32(
                false, a[k], false, bb, (short)0, acc, false, false);
        }

        const float gx0 = gt_bboxes[(b * N_ + n) * 4 + 0];
        const float gy0 = gt_bboxes[(b * N_ + n) * 4 + 1];
        const float gx1 = gt_bboxes[(b * N_ + n) * 4 + 2];
        const float gy1 = gt_bboxes[(b * N_ + n) * 4 + 3];
        const float mg  = mask_gt[b * N_ + n];
        const float ag  = (gx1 - gx0) * (gy1 - gy0);

#pragma unroll
        for (int r = 0; r < 8; ++r) {
            const int hw = hwl + r;
            const float ax  = anc[hw * 2 + 0];
            const float ay  = anc[hw * 2 + 1];
            const float px0 = pd_bboxes[((size_t)b * HW_ + hw) * 4 + 0];
            const float py0 = pd_bboxes[((size_t)b * HW_ + hw) * 4 + 1];
            const float px1 = pd_bboxes[((size_t)b * HW_ + hw) * 4 + 2];
            const float py1 = pd_bboxes[((size_t)b * HW_ + hw) * 4 + 3];

            const float mind  = fminf(fminf(ax - gx0, ay - gy0),
                                      fminf(gx1 - ax, gy1 - ay));
            const bool  valid = (mind > EPSF) && (mg > 0.0f);

            const float ltx = fmaxf(gx0, px0), lty = fmaxf(gy0, py0);
            const float rbx = fminf(gx1, px1), rby = fminf(gy1, py1);
            const float w   = fmaxf(rbx - ltx, 0.0f);
            const float h   = fmaxf(rby - lty, 0.0f);
            const float inter = w * h;
            const float ap    = (px1 - px0) * (py1 - py0);
            const float iou   = inter / (ag + ap - inter + EPSF);

            const float ov = valid ? iou    : 0.0f;
            const float s  = valid ? acc[r] : 0.0f;
            const float o3 = ov * ov * ov;
            const float am = s * o3 * o3;          // score^1 * iou^6

            const size_t idx = ((size_t)b * HW_ + hw) * N_ + n;
            OV[idx] = ov;
            AM[idx] = am;
        }
    }
}

// ---------------------------------------------------------------------------
// Top-13 per (b,n) row with lax.top_k tie semantics (ties -> lowest index),
// then scatter-select: cnt += 1, seln = n, gated by in-gts recheck.
// One wave per row.
// ---------------------------------------------------------------------------
__global__ __launch_bounds__(32)
void k_topk(const float* __restrict__ AM,
            const float* __restrict__ anc,
            const float* __restrict__ gt_bboxes,
            const float* __restrict__ mask_gt,
            unsigned* __restrict__ cnt, int* __restrict__ seln) {
    const int bn = blockIdx.x;
    if (mask_gt[bn] <= 0.0f) return;          // topk_mask false -> no selections
    const int b = bn / N_, n = bn % N_;
    const int lane = threadIdx.x;

    float lv[TK_]; int li[TK_];
#pragma unroll
    for (int s = 0; s < TK_; ++s) { lv[s] = -1.0f; li[s] = 0x7fffffff; }
    float minv = -1.0f; int minp = 0;

    for (int i = lane; i < HW_; i += 32) {
        const float v = AM[((size_t)b * HW_ + i) * N_ + n];
        if (v > minv) {                        // strict: keeps earlier (lower idx) ties
            lv[minp] = v; li[minp] = i;
            // lex-worst slot: smallest value, then LARGEST index
            minv = lv[0]; minp = 0;
#pragma unroll
            for (int s = 1; s < TK_; ++s)
                if (lv[s] < minv || (lv[s] == minv && li[s] > li[minp])) {
                    minv = lv[s]; minp = s;
                }
        }
    }

    const float gx0 = gt_bboxes[bn * 4 + 0];
    const float gy0 = gt_bboxes[bn * 4 + 1];
    const float gx1 = gt_bboxes[bn * 4 + 2];
    const float gy1 = gt_bboxes[bn * 4 + 3];

    for (int t = 0; t < TK_; ++t) {
        // lane-local best remaining (value desc, index asc)
        float bv = -2.0f; int bi = 0x7fffffff;
#pragma unroll
        for (int s = 0; s < TK_; ++s)
            if (lv[s] > bv || (lv[s] == bv && li[s] < bi)) { bv = lv[s]; bi = li[s]; }

        // wave argmax (lexicographic), converges on all lanes
        float v = bv; int ix = bi;
#pragma unroll
        for (int off = 16; off >= 1; off >>= 1) {
            const float ovv = __shfl_xor(v, off, 32);
            const int   oi  = __shfl_xor(ix, off, 32);
            if (ovv > v || (ovv == v && oi < ix)) { v = ovv; ix = oi; }
        }
        // remove from owner (hw indices are unique across lanes)
#pragma unroll
        for (int s = 0; s < TK_; ++s)
            if (li[s] == ix) { lv[s] = -2.0f; li[s] = 0x7fffffff; }

        if (lane == 0) {
            const int hw = ix;
            const float ax = anc[hw * 2], ay = anc[hw * 2 + 1];
            const float mind = fminf(fminf(ax - gx0, ay - gy0),
                                     fminf(gx1 - ax, gy1 - ay));
            if (mind > EPSF) {                 // mask_in_gts gate
                const int a = b * HW_ + hw;
                atomicAdd(&cnt[a], 1u);
                atomicExch(&seln[a], n);
            }
        }
    }
}

// ---------------------------------------------------------------------------
// Per-anchor conflict resolution + pos_align/pos_overlap atomic maxima.
// ---------------------------------------------------------------------------
__global__ void k_resolve(const float* __restrict__ OV, const float* __restrict__ AM,
                          const unsigned* __restrict__ cnt, const int* __restrict__ seln,
                          int* __restrict__ tg, int* __restrict__ fgm,
                          unsigned* __restrict__ pal, unsigned* __restrict__ pov) {
    const int a = blockIdx.x * blockDim.x + threadIdx.x;
    if (a >= B_ * HW_) return;
    const int b = a / HW_;
    const unsigned c = cnt[a];
    int t = 0, f = 0;
    if (c == 1u) { t = seln[a]; f = 1; }
    else if (c > 1u) {
        // is_max replacement: first-occurrence argmax over masked overlaps column
        const float* col = OV + (size_t)a * N_;
        float best = col[0]; t = 0;
        for (int n = 1; n < N_; ++n) {
            const float v = col[n];
            if (v > best) { best = v; t = n; }
        }
        f = 1;
    }
    tg[a] = t; fgm[a] = f;
    if (f) {
        const float av = AM[(size_t)a * N_ + t];
        const float ov = OV[(size_t)a * N_ + t];
        // values are >= 0: uint bit order == float order
        atomicMax(&pal[b * N_ + t], __float_as_uint(av));
        atomicMax(&pov[b * N_ + t], __float_as_uint(ov));
    }
}

// ---------------------------------------------------------------------------
// Output pass 1: labels, bboxes, fg, tgidx, and per-anchor norm.
// ---------------------------------------------------------------------------
__global__ void k_out1(const float* __restrict__ AM,
                       const int* __restrict__ gt_labels,
                       const float* __restrict__ gt_bboxes,
                       const int* __restrict__ tg, const int* __restrict__ fgm,
                       const unsigned* __restrict__ pal, const unsigned* __restrict__ pov,
                       int* __restrict__ labv, float* __restrict__ normv,
                       float* __restrict__ out) {
    const int a = blockIdx.x * blockDim.x + threadIdx.x;
    if (a >= B_ * HW_) return;
    const int b = a / HW_;
    const int t = tg[a], f = fgm[a];
    int lbl = gt_labels[b * N_ + t];
    lbl = lbl < 0 ? 0 : lbl;                   // clip(., 0)
    float nv = 0.0f;
    if (f) {
        const float av = AM[(size_t)a * N_ + t];
        const float po = __uint_as_float(pov[b * N_ + t]);
        const float pa = __uint_as_float(pal[b * N_ + t]);
        nv = av * po / (pa + EPSF);
    }
    labv[a] = lbl; normv[a] = nv;

    float* o_lab = out;
    float* o_box = out + (size_t)B_ * HW_;
    float* o_fg  = out + (size_t)B_ * HW_ * (size_t)(1 + 4 + C_);
    float* o_tg  = o_fg + (size_t)B_ * HW_;

    o_lab[a] = (float)lbl;
    const float* gb = gt_bboxes + (size_t)(b * N_ + t) * 4;
    o_box[(size_t)a * 4 + 0] = gb[0];
    o_box[(size_t)a * 4 + 1] = gb[1];
    o_box[(size_t)a * 4 + 2] = gb[2];
    o_box[(size_t)a * 4 + 3] = gb[3];
    o_fg[a] = f ? 1.0f : 0.0f;
    o_tg[a] = (float)t;
}

// ---------------------------------------------------------------------------
// Output pass 2: coalesced target_scores = one_hot(label) * norm.
// ---------------------------------------------------------------------------
__global__ void k_out2(const int* __restrict__ labv, const float* __restrict__ normv,
                       float* __restrict__ o_sc) {
    const size_t i = (size_t)blockIdx.x * blockDim.x + threadIdx.x;
    if (i >= (size_t)B_ * HW_ * C_) return;
    const int a = (int)(i / C_);
    const int c = (int)(i % C_);
    o_sc[i] = (c == labv[a]) ? normv[a] : 0.0f;
}

// ---------------------------------------------------------------------------
extern "C" void kernel_launch(void* const* d_in, const int* in_sizes, int n_in,
                              void* d_out, int out_size, void* d_ws, size_t ws_size,
                              hipStream_t stream) {
    const float* pd_scores = (const float*)d_in[0];
    const float* pd_bboxes = (const float*)d_in[1];
    const float* anc       = (const float*)d_in[2];
    const int*   gt_labels = (const int*)d_in[3];
    const float* gt_bboxes = (const float*)d_in[4];
    const float* mask_gt   = (const float*)d_in[5];
    float* out = (float*)d_out;

    const size_t PAIRS = (size_t)B_ * HW_ * N_;
    const size_t BH    = (size_t)B_ * HW_;

    float*    OV    = (float*)d_ws;
    float*    AM    = OV + PAIRS;
    unsigned* cnt   = (unsigned*)(AM + PAIRS);
    int*      seln  = (int*)(cnt + BH);
    unsigned* pal   = (unsigned*)(seln + BH);
    unsigned* pov   = pal + (size_t)B_ * N_;
    int*      tg    = (int*)(pov + (size_t)B_ * N_);
    int*      fgm   = tg + BH;
    int*      labv  = fgm + BH;
    float*    normv = (float*)(labv + BH);

    k_init<<<(B_ * HW_ + 255) / 256, 256, 0, stream>>>(cnt, seln, pal, pov);

    dim3 g1(HW_ / 16, B_);
    k_metric<<<g1, 32, 0, stream>>>(pd_scores, pd_bboxes, anc, gt_labels,
                                    gt_bboxes, mask_gt, OV, AM);

    k_topk<<<B_ * N_, 32, 0, stream>>>(AM, anc, gt_bboxes, mask_gt, cnt, seln);

    k_resolve<<<(B_ * HW_ + 255) / 256, 256, 0, stream>>>(OV, AM, cnt, seln,
                                                          tg, fgm, pal, pov);

    k_out1<<<(B_ * HW_ + 255) / 256, 256, 0, stream>>>(AM, gt_labels, gt_bboxes,
                                                       tg, fgm, pal, pov,
                                                       labv, normv, out);

    float* o_sc = out + (size_t)B_ * HW_ * 5;   // labels(1)+bbox(4) before scores
    const size_t nsc = (size_t)B_ * HW_ * C_;
    k_out2<<<(unsigned)((nsc + 255) / 256), 256, 0, stream>>>(labv, normv, o_sc);
}